// GraphFeatureExtractor_50586124812938
// MI455X (gfx1250) — compile-verified
//
#include <hip/hip_runtime.h>
#include <math.h>

// ---------------------------------------------------------------------------
// GAT feature extractor for MI455X (gfx1250, wave32).
// Dense GEMMs -> V_WMMA_F32_16X16X4_F32. Edge phase -> 3-pass scatter-softmax
// with L2-resident node arrays (total ~46MB << 192MB L2).
// ---------------------------------------------------------------------------

typedef __attribute__((ext_vector_type(2))) float v2f;
typedef __attribute__((ext_vector_type(8))) float v8f;

#define NEG_SLOPE 0.2f
static constexpr unsigned ORD_NEG_INF = 0x007FFFFFu; // ordered-key of -inf

__device__ __forceinline__ unsigned ordf(float f) {
  unsigned u = __float_as_uint(f);
  return (u & 0x80000000u) ? ~u : (u | 0x80000000u);
}
__device__ __forceinline__ float unordf(unsigned k) {
  return __uint_as_float((k & 0x80000000u) ? (k & 0x7fffffffu) : ~k);
}
__device__ __forceinline__ float lrelu(float x) { return x > 0.0f ? x : NEG_SLOPE * x; }
__device__ __forceinline__ float mishf(float x) {
  float sp = (x > 20.0f) ? x : log1pf(__expf(x));
  return x * tanhf(sp);
}

// ------------------------------- fills -------------------------------------
__global__ void fill_f32(float* p, float v, int n) {
  for (int i = blockIdx.x * blockDim.x + threadIdx.x; i < n; i += gridDim.x * blockDim.x)
    p[i] = v;
}
__global__ void fill_u32(unsigned* p, unsigned v, int n) {
  for (int i = blockIdx.x * blockDim.x + threadIdx.x; i < n; i += gridDim.x * blockDim.x)
    p[i] = v;
}

// ------------------------- WMMA GEMMs (fp32) -------------------------------
// H[N,16] = X[N,128] @ W[128,16].  One wave per 16-row tile, 32 WMMA steps.
__global__ void gemm_n128x16(const float* __restrict__ X, const float* __restrict__ W,
                             float* __restrict__ H, int nNodes) {
  int wave = (blockIdx.x * blockDim.x + threadIdx.x) >> 5;
  int lane = threadIdx.x & 31;
  int row0 = wave << 4;
  if (row0 >= nNodes) return;
  int m = lane & 15;
  int hi = lane >> 4; // lane half: 0 -> K0/K1 rows 0..7, 1 -> K2/K3 rows 8..15
  const float* xr = X + (size_t)(row0 + m) * 128;
  v8f c = {0.f, 0.f, 0.f, 0.f, 0.f, 0.f, 0.f, 0.f};
#pragma unroll
  for (int k = 0; k < 128; k += 4) {
    v2f a, b;
    a.x = xr[k + 2 * hi];
    a.y = xr[k + 2 * hi + 1];
    b.x = W[(k + 2 * hi) * 16 + m];
    b.y = W[(k + 2 * hi + 1) * 16 + m];
    c = __builtin_amdgcn_wmma_f32_16x16x4_f32(false, a, false, b, (short)0, c, false, false);
  }
#pragma unroll
  for (int r = 0; r < 8; r++)
    H[(size_t)(row0 + r + 8 * hi) * 16 + m] = c[r];
}

// H[N,32] = X[N,16] @ W[16,32].  One wave per 16-row tile, 2 col-tiles share A.
__global__ void gemm_n16x32(const float* __restrict__ X, const float* __restrict__ W,
                            float* __restrict__ H, int nNodes) {
  int wave = (blockIdx.x * blockDim.x + threadIdx.x) >> 5;
  int lane = threadIdx.x & 31;
  int row0 = wave << 4;
  if (row0 >= nNodes) return;
  int m = lane & 15;
  int hi = lane >> 4;
  const float* xr = X + (size_t)(row0 + m) * 16;
  v8f c0 = {0.f, 0.f, 0.f, 0.f, 0.f, 0.f, 0.f, 0.f};
  v8f c1 = {0.f, 0.f, 0.f, 0.f, 0.f, 0.f, 0.f, 0.f};
#pragma unroll
  for (int k = 0; k < 16; k += 4) {
    v2f a, b0, b1;
    a.x = xr[k + 2 * hi];
    a.y = xr[k + 2 * hi + 1];
    b0.x = W[(k + 2 * hi) * 32 + m];
    b0.y = W[(k + 2 * hi + 1) * 32 + m];
    b1.x = W[(k + 2 * hi) * 32 + 16 + m];
    b1.y = W[(k + 2 * hi + 1) * 32 + 16 + m];
    c0 = __builtin_amdgcn_wmma_f32_16x16x4_f32(false, a, false, b0, (short)0, c0, false, false);
    c1 = __builtin_amdgcn_wmma_f32_16x16x4_f32(false, a, false, b1, (short)0, c1, false, false);
  }
#pragma unroll
  for (int r = 0; r < 8; r++) {
    H[(size_t)(row0 + r + 8 * hi) * 32 + m] = c0[r];
    H[(size_t)(row0 + r + 8 * hi) * 32 + 16 + m] = c1[r];
  }
}

// ------------------------- per-node attention dots -------------------------
__global__ void alpha_h4c4(const float* __restrict__ H, const float* __restrict__ aS,
                           const float* __restrict__ aD, float* __restrict__ as_,
                           float* __restrict__ ad_, int n) {
  int i = blockIdx.x * blockDim.x + threadIdx.x;
  if (i >= n) return;
  const float* hr = H + (size_t)i * 16;
#pragma unroll
  for (int h = 0; h < 4; h++) {
    float s = 0.f, d = 0.f;
#pragma unroll
    for (int c = 0; c < 4; c++) {
      float v = hr[h * 4 + c];
      s += v * aS[h * 4 + c];
      d += v * aD[h * 4 + c];
    }
    as_[i * 4 + h] = s;
    ad_[i * 4 + h] = d;
  }
}

__global__ void alpha_h1c32(const float* __restrict__ H, const float* __restrict__ aS,
                            const float* __restrict__ aD, float* __restrict__ as_,
                            float* __restrict__ ad_, int n) {
  int i = blockIdx.x * blockDim.x + threadIdx.x;
  if (i >= n) return;
  const float* hr = H + (size_t)i * 32;
  float s = 0.f, d = 0.f;
#pragma unroll
  for (int c = 0; c < 32; c++) {
    float v = hr[c];
    s += v * aS[c];
    d += v * aD[c];
  }
  as_[i] = s;
  ad_[i] = d;
}

// ------------------------- edge passes (scatter softmax) -------------------
// Edge ids >= nE are the self-loops (src = dst = e - nE).
template <int HDS>
__global__ void edge_max_k(const int* __restrict__ src, const int* __restrict__ dst,
                           const float* __restrict__ as_, const float* __restrict__ ad_,
                           unsigned* __restrict__ emax, int nE, int nNodes) {
  int e = blockIdx.x * blockDim.x + threadIdx.x;
  if (e >= nE + nNodes) return;
  int s, d;
  if (e < nE) { s = src[e]; d = dst[e]; } else { s = d = e - nE; }
#pragma unroll
  for (int h = 0; h < HDS; h++) {
    float v = lrelu(as_[s * HDS + h] + ad_[d * HDS + h]);
    atomicMax(&emax[d * HDS + h], ordf(v));
  }
}

template <int HDS>
__global__ void edge_den_k(const int* __restrict__ src, const int* __restrict__ dst,
                           const float* __restrict__ as_, const float* __restrict__ ad_,
                           const unsigned* __restrict__ emax, float* __restrict__ den,
                           int nE, int nNodes) {
  int e = blockIdx.x * blockDim.x + threadIdx.x;
  if (e >= nE + nNodes) return;
  int s, d;
  if (e < nE) { s = src[e]; d = dst[e]; } else { s = d = e - nE; }
#pragma unroll
  for (int h = 0; h < HDS; h++) {
    float v = lrelu(as_[s * HDS + h] + ad_[d * HDS + h]);
    float m = unordf(emax[d * HDS + h]);
    atomicAdd(&den[d * HDS + h], __expf(v - m));
  }
}

template <int HDS, int CH>
__global__ void edge_agg_k(const int* __restrict__ src, const int* __restrict__ dst,
                           const float* __restrict__ as_, const float* __restrict__ ad_,
                           const unsigned* __restrict__ emax, const float* __restrict__ den,
                           const float* __restrict__ H, float* __restrict__ agg,
                           int nE, int nNodes) {
  int e = blockIdx.x * blockDim.x + threadIdx.x;
  if (e >= nE + nNodes) return;
  int s, d;
  if (e < nE) { s = src[e]; d = dst[e]; } else { s = d = e - nE; }
  const float* hr = H + (size_t)s * (HDS * CH);
  __builtin_prefetch(hr, 0, 0); // global_prefetch_b8 on the gathered feature row
#pragma unroll
  for (int h = 0; h < HDS; h++) {
    float v = lrelu(as_[s * HDS + h] + ad_[d * HDS + h]);
    float m = unordf(emax[d * HDS + h]);
    float alpha = __expf(v - m) / (den[d * HDS + h] + 1e-16f);
#pragma unroll
    for (int c = 0; c < CH; c++)
      atomicAdd(&agg[(size_t)d * (HDS * CH) + h * CH + c], hr[h * CH + c] * alpha);
  }
}

// ------------------------- bias + mish (in place) --------------------------
__global__ void mish_bias(float* __restrict__ A, const float* __restrict__ b, int C, int total) {
  for (int i = blockIdx.x * blockDim.x + threadIdx.x; i < total; i += gridDim.x * blockDim.x) {
    float x = A[i] + b[i % C];
    A[i] = mishf(x);
  }
}

// ------------------------- global mean pool --------------------------------
__global__ void pool_sum(const float* __restrict__ H, const int* __restrict__ batch,
                         float* __restrict__ out, float* __restrict__ cnts, int n) {
  __shared__ float sh[64 * 32];
  __shared__ float shc[64];
  for (int i = threadIdx.x; i < 64 * 32; i += blockDim.x) sh[i] = 0.f;
  for (int i = threadIdx.x; i < 64; i += blockDim.x) shc[i] = 0.f;
  __syncthreads();
  int node = blockIdx.x * blockDim.x + threadIdx.x;
  if (node < n) {
    int g = batch[node];
    const float* hr = H + (size_t)node * 32;
#pragma unroll
    for (int c = 0; c < 32; c++) atomicAdd(&sh[g * 32 + c], hr[c]);
    atomicAdd(&shc[g], 1.0f);
  }
  __syncthreads();
  for (int i = threadIdx.x; i < 64 * 32; i += blockDim.x) {
    float v = sh[i];
    if (v != 0.f) atomicAdd(&out[i], v);
  }
  for (int i = threadIdx.x; i < 64; i += blockDim.x) {
    float v = shc[i];
    if (v != 0.f) atomicAdd(&cnts[i], v);
  }
}

__global__ void pool_div(float* __restrict__ out, const float* __restrict__ cnts) {
  int i = blockIdx.x * blockDim.x + threadIdx.x;
  if (i < 64 * 32) out[i] /= fmaxf(cnts[i >> 5], 1.0f);
}

// ---------------------------------------------------------------------------
extern "C" void kernel_launch(void* const* d_in, const int* in_sizes, int n_in,
                              void* d_out, int out_size, void* d_ws, size_t ws_size,
                              hipStream_t stream) {
  const float* x     = (const float*)d_in[0];
  const int*   ei    = (const int*)d_in[1];
  const int*   batch = (const int*)d_in[2];
  const float* W1    = (const float*)d_in[3];
  const float* b1    = (const float*)d_in[4];
  const float* aS1   = (const float*)d_in[5];
  const float* aD1   = (const float*)d_in[6];
  const float* W2    = (const float*)d_in[7];
  const float* b2    = (const float*)d_in[8];
  const float* aS2   = (const float*)d_in[9];
  const float* aD2   = (const float*)d_in[10];

  const int N = in_sizes[0] / 128;
  const int E = in_sizes[1] / 2;
  const int* src = ei;
  const int* dst = ei + E;
  float* out = (float*)d_out;

  // ---- workspace carve-out (all node arrays; ~46MB total -> L2 resident)
  float* p = (float*)d_ws;
  float*    h1    = p;               p += (size_t)16 * N;
  float*    as1   = p;               p += (size_t)4 * N;
  float*    ad1   = p;               p += (size_t)4 * N;
  unsigned* emax1 = (unsigned*)p;    p += (size_t)4 * N;
  float*    den1  = p;               p += (size_t)4 * N;
  float*    agg1  = p;               p += (size_t)16 * N; // becomes layer-2 input after mish
  float*    h3    = p;               p += (size_t)32 * N;
  float*    as2   = p;               p += (size_t)N;
  float*    ad2   = p;               p += (size_t)N;
  unsigned* emax2 = (unsigned*)p;    p += (size_t)N;
  float*    den2  = p;               p += (size_t)N;
  float*    agg2  = p;               p += (size_t)32 * N; // final node features after mish
  float*    cnts  = p;               p += 64;

  const int BLK = 256;
  auto grid = [](long long n) { return (int)((n + 255) / 256); };

  // ---- init accumulators (fresh every call: graph-replay safe)
  fill_u32<<<grid(4LL * N), BLK, 0, stream>>>(emax1, ORD_NEG_INF, 4 * N);
  fill_f32<<<grid(4LL * N), BLK, 0, stream>>>(den1, 0.f, 4 * N);
  fill_f32<<<grid(16LL * N), BLK, 0, stream>>>(agg1, 0.f, 16 * N);
  fill_u32<<<grid(N), BLK, 0, stream>>>(emax2, ORD_NEG_INF, N);
  fill_f32<<<grid(N), BLK, 0, stream>>>(den2, 0.f, N);
  fill_f32<<<grid(32LL * N), BLK, 0, stream>>>(agg2, 0.f, 32 * N);
  fill_f32<<<grid(64 * 32), BLK, 0, stream>>>(out, 0.f, 64 * 32);
  fill_f32<<<1, 64, 0, stream>>>(cnts, 0.f, 64);

  const int nWaves = (N + 15) / 16;
  const int gemmBlocks = (nWaves * 32 + BLK - 1) / BLK;
  const long long nTot = (long long)E + N; // edges + self loops

  // ---- layer 1 (heads=4, out_ch=4)
  gemm_n128x16<<<gemmBlocks, BLK, 0, stream>>>(x, W1, h1, N);
  alpha_h4c4<<<grid(N), BLK, 0, stream>>>(h1, aS1, aD1, as1, ad1, N);
  edge_max_k<4><<<grid(nTot), BLK, 0, stream>>>(src, dst, as1, ad1, emax1, E, N);
  edge_den_k<4><<<grid(nTot), BLK, 0, stream>>>(src, dst, as1, ad1, emax1, den1, E, N);
  edge_agg_k<4, 4><<<grid(nTot), BLK, 0, stream>>>(src, dst, as1, ad1, emax1, den1, h1, agg1, E, N);
  mish_bias<<<grid(16LL * N), BLK, 0, stream>>>(agg1, b1, 16, 16 * N);

  // ---- layer 2 (heads=1, out_ch=32)
  gemm_n16x32<<<gemmBlocks, BLK, 0, stream>>>(agg1, W2, h3, N);
  alpha_h1c32<<<grid(N), BLK, 0, stream>>>(h3, aS2, aD2, as2, ad2, N);
  edge_max_k<1><<<grid(nTot), BLK, 0, stream>>>(src, dst, as2, ad2, emax2, E, N);
  edge_den_k<1><<<grid(nTot), BLK, 0, stream>>>(src, dst, as2, ad2, emax2, den2, E, N);
  edge_agg_k<1, 32><<<grid(nTot), BLK, 0, stream>>>(src, dst, as2, ad2, emax2, den2, h3, agg2, E, N);
  mish_bias<<<grid(32LL * N), BLK, 0, stream>>>(agg2, b2, 32, 32 * N);

  // ---- global mean pool
  pool_sum<<<grid(N), BLK, 0, stream>>>(agg2, batch, out, cnts, N);
  pool_div<<<grid(64 * 32), BLK, 0, stream>>>(out, cnts);
}